// TSM_New_33535104647443
// MI455X (gfx1250) — compile-verified
//
#include <hip/hip_runtime.h>
#include <stdint.h>

// ---------------------------------------------------------------------------
// TSM temporal-shift (B,T,C,H,W) f32.  Pure data movement:
//   ~616 MB written, ~411 MB read  ->  ~44 us roofline at 23.3 TB/s.
// Copy planes use the CDNA5 async global<->LDS DMA path (ASYNCcnt-tracked,
// no VGPR staging) with NT cache policy (stream >> 192 MB L2, zero reuse).
// Zero planes are write-only NT vector stores.
// One 256-thread block per output HW plane (12288 blocks, 53 KB LDS each).
// ---------------------------------------------------------------------------

typedef float v4f __attribute__((ext_vector_type(4)));

#define THREADS    256
#define MAX_CHUNKS 13   // ceil((112*112/4) / 256) float4-chunks per thread
#define OVL        7    // chunks stored in first burst (overlap with tail loads)

// gfx1250 async copy: global -> LDS, 16B per active lane, tracked by ASYNCcnt.
//   global_load_async_to_lds_b128  vdst(LDS addr), vaddr(32b offset), saddr
__device__ __forceinline__ void async_load_b128(uint32_t lds_addr, uint32_t voff,
                                                const void* sbase) {
  asm volatile("global_load_async_to_lds_b128 %0, %1, %2 th:TH_LOAD_NT"
               :
               : "v"(lds_addr), "v"(voff), "s"(sbase)
               : "memory");
}

// gfx1250 async copy: LDS -> global, 16B per active lane, tracked by ASYNCcnt.
//   global_store_async_from_lds_b128  vaddr(32b offset), vsrc(LDS addr), saddr
__device__ __forceinline__ void async_store_b128(uint32_t voff, uint32_t lds_addr,
                                                 const void* sbase) {
  asm volatile("global_store_async_from_lds_b128 %0, %1, %2 th:TH_STORE_NT"
               :
               : "v"(voff), "v"(lds_addr), "s"(sbase)
               : "memory");
}

// s_wait_asynccnt takes a SIMM16 literal -> templated immediate.
template <int N>
__device__ __forceinline__ void wait_async() {
  asm volatile("s_wait_asynccnt %0" :: "i"(N) : "memory");
}

__global__ __launch_bounds__(THREADS)
void tsm_shift_async_kernel(const float* __restrict__ x,
                            const float* __restrict__ shiftp,
                            const int*   __restrict__ elemp,
                            float* __restrict__ out,
                            int T, int C, int HW) {
  // Per-block staging buffer: 256 threads * 13 * 16B = 53,248 B of LDS.
  __shared__ v4f smem[THREADS * MAX_CHUNKS];

  const int plane = blockIdx.x;          // plane id over B*T*C
  const int c = plane % C;
  const int t = (plane / C) % T;
  const int b = plane / (C * T);

  const int   e  = elemp[0];             // elements (3)
  const float sf = shiftp[0];            // shift_factor (0.25)
  const int   k  = (int)floorf((float)T * sf);

  // Replicate reference index-set construction (with trim rules):
  //   idx_for  = arange(0,C,e); drop last if last >= C-1
  //   idx_back = idx_for + 1;   drop last if last >= C-1
  const int  lastf  = ((C - 1) / e) * e;
  const bool trimf  = (lastf >= C - 1);
  const bool is_for = ((c % e) == 0) && !(trimf && (c == lastf));
  const int  cb     = c - 1;
  const bool in_for = (cb >= 0) && ((cb % e) == 0) && !(trimf && (cb == lastf));
  const int  lastb  = (trimf ? (lastf - e) : lastf) + 1;
  const bool trimb  = (lastb >= C - 1);
  const bool is_back = in_for && !(trimb && (c == lastb));

  // Reference write order: idx_back updates fully override idx_for updates.
  bool zero; int src_t;
  if (is_back)     { zero = (t < k);     src_t = t - k; }
  else if (is_for) { zero = (t < T - k); src_t = t;     }
  else             { zero = false;       src_t = t;     }

  float* __restrict__ dst = out + (size_t)plane * (size_t)HW;
  const int NV4 = HW >> 2;
  const int tid = threadIdx.x;

  if (zero) {
    // Write-only plane: NT vector stores, no reads at all.
    const v4f z = {0.f, 0.f, 0.f, 0.f};
    v4f* __restrict__ d4 = (v4f*)dst;
    for (int i = tid; i < NV4; i += THREADS)
      __builtin_nontemporal_store(z, d4 + i);
    for (int i = (NV4 << 2) + tid; i < HW; i += THREADS)   // tail (unused here)
      __builtin_nontemporal_store(0.0f, dst + i);
    return;
  }

  const float* __restrict__ src =
      x + ((size_t)(b * T + src_t) * (size_t)C + (size_t)c) * (size_t)HW;

  const uint32_t lds_base = (uint32_t)(uintptr_t)(&smem[0]);

  // Phase 1: burst all async global->LDS copies for this thread's chunks
  // (up to 13 in flight per wave; ~53 KB in flight per block).
  int nc = 0;
  for (int i = tid; i < NV4 && nc < MAX_CHUNKS; i += THREADS, ++nc) {
    const uint32_t off16 = (uint32_t)i * 16u;               // byte offset
    async_load_b128(lds_base + off16, off16, (const void*)src);
  }

  // Phase 2a: ASYNC loads complete in order, so after n loads,
  // s_wait_asynccnt (n - OVL) guarantees chunks [0, OVL) are in LDS.
  // Store them while the remaining loads are still in flight.
  // (nc is wave-uniform: 13 for waves 0-1, 12 for waves 2-7 at HW=12544.)
  if (nc == MAX_CHUNKS)          wait_async<MAX_CHUNKS - OVL>();      // 6
  else if (nc == MAX_CHUNKS - 1) wait_async<MAX_CHUNKS - 1 - OVL>();  // 5
  else                           wait_async<0>();                      // generic

  int q = 0;
  for (int i = tid; q < nc && q < OVL; i += THREADS, ++q) {
    const uint32_t off16 = (uint32_t)i * 16u;
    async_store_b128(off16, lds_base + off16, (const void*)dst);
  }

  // Phase 2b: drain (ASYNCcnt mixes load/store completions, so only <=0 is
  // safe here), then store the remaining chunks; s_endpgm's implicit
  // wait-idle retires them.
  wait_async<0>();
  for (int i = tid + OVL * THREADS; q < nc; i += THREADS, ++q) {
    const uint32_t off16 = (uint32_t)i * 16u;
    async_store_b128(off16, lds_base + off16, (const void*)dst);
  }

  // Scalar tail if HW % 4 != 0 (not hit for 112*112).
  for (int i = (NV4 << 2) + tid; i < HW; i += THREADS)
    dst[i] = __builtin_nontemporal_load(src + i);
}

extern "C" void kernel_launch(void* const* d_in, const int* in_sizes, int n_in,
                              void* d_out, int out_size, void* d_ws, size_t ws_size,
                              hipStream_t stream) {
  const float* x  = (const float*)d_in[0];   // (B,16,96,112,112) f32
  const float* sf = (const float*)d_in[1];   // shift_factor scalar
  const int*   el = (const int*)d_in[2];     // elements scalar
  float* out = (float*)d_out;

  const int T = 16, C = 96, H = 112, W = 112;
  const int HW = H * W;
  const int B  = in_sizes[0] / (T * C * HW); // 8 for the reference shapes

  dim3 grid((unsigned)(B * T * C));          // one block per output plane
  tsm_shift_async_kernel<<<grid, THREADS, 0, stream>>>(x, sf, el, out, T, C, HW);
}